// MahalanobisLoss2D_17214228923026
// MI455X (gfx1250) — compile-verified
//
#include <hip/hip_runtime.h>
#include <stdint.h>

// ---------------------------------------------------------------------------
// AR(1) Mahalanobis loss, MI455X (gfx1250).
// Bandwidth-bound: 268 MB streamed @ 23.3 TB/s => ~11.5 us floor.
// Data path: TDM tensor_load_to_lds DMA (global -> LDS), ds_load_b128 reads,
// wave32 shuffle reductions. No matrix structure => no WMMA by design.
// ---------------------------------------------------------------------------

#if __has_include(<hip/amd_detail/amd_gfx1250_TDM.h>)
#define ATH_TDM_SIX_ARGS 1   // amdgpu-toolchain / clang-23: 6-arg builtin
#endif

typedef unsigned int u32x4 __attribute__((ext_vector_type(4)));
typedef int          i32x4 __attribute__((ext_vector_type(4)));
typedef int          i32x8 __attribute__((ext_vector_type(8)));

#define ATH_D 4096
#define ATH_THREADS 256
#define ATH_PER_THREAD (ATH_D / ATH_THREADS)   // 16 columns per thread

// Issue one TDM descriptor: copy `nelem` contiguous f32 from global to LDS.
// D# group0: count=1 | lds_addr | global_addr[56:0] | type=2 ("image").
// D# group1: data_size=4B, tensor_dim0=tile_dim0=nelem, tensor_dim1=tile_dim1=1,
//            tensor_dim0_stride=nelem (dense row), no padding/iterate/multicast.
__device__ __forceinline__ void tdm_load_row_f32(uint32_t lds_byte_addr,
                                                 const float* gsrc, int nelem) {
  uint64_t ga = (uint64_t)(uintptr_t)gsrc;
  u32x4 g0;
  g0[0] = 1u;                                                   // count=1, user mode
  g0[1] = lds_byte_addr;                                        // LDS dest (bytes)
  g0[2] = (uint32_t)(ga & 0xFFFFFFFFull);                       // global_addr[31:0]
  g0[3] = (uint32_t)((ga >> 32) & 0x01FFFFFFull) | (2u << 30);  // ga[56:32] | type=2
  i32x8 g1;
  g1[0] = (int)(2u << 16);                  // wg_mask=0, data_size=2 (4 bytes)
  g1[1] = (int)((uint32_t)nelem << 16);     // tensor_dim0[15:0] @ bits 63:48
  g1[2] = (int)(1u << 16);                  // tensor_dim0 hi=0 | tensor_dim1=1
  g1[3] = (int)((uint32_t)nelem << 16);     // tensor_dim1 hi=0 | tile_dim0=nelem
  g1[4] = 1;                                // tile_dim1=1, tile_dim2=0
  g1[5] = nelem;                            // tensor_dim0_stride[31:0]
  g1[6] = 0;                                // stride hi, dim1_stride lo
  g1[7] = 0;                                // dim1_stride hi
  i32x4 z4 = {0, 0, 0, 0};
#ifdef ATH_TDM_SIX_ARGS
  i32x8 z8 = {0, 0, 0, 0, 0, 0, 0, 0};
  __builtin_amdgcn_tensor_load_to_lds(g0, g1, z4, z4, z8, 0);
#else
  __builtin_amdgcn_tensor_load_to_lds(g0, g1, z4, z4, 0);
#endif
}

__global__ __launch_bounds__(ATH_THREADS) void
ar1_row_quad_kernel(const float* __restrict__ y_true,
                    const float* __restrict__ y_pred,
                    const float* __restrict__ param,
                    const int*   __restrict__ n,
                    float*       __restrict__ rowq) {
  __shared__ __align__(16) float s_yt[ATH_D];
  __shared__ __align__(16) float s_yp[ATH_D];
  __shared__ float s_red[ATH_THREADS / 32];

  const int row = blockIdx.x;
  const int tid = threadIdx.x;

  // Wave 0 issues both row DMAs. Branch is wave-uniform (threads 0..31),
  // so EXEC is all-ones inside; TDM ignores EXEC anyway. TENSORcnt is
  // per-wave, so the issuing wave waits, then the barrier releases everyone.
  if (tid < 32) {
    tdm_load_row_f32((uint32_t)(uintptr_t)(void*)s_yt,
                     y_true + (size_t)row * ATH_D, ATH_D);
    tdm_load_row_f32((uint32_t)(uintptr_t)(void*)s_yp,
                     y_pred + (size_t)row * ATH_D, ATH_D);
    __builtin_amdgcn_s_wait_tensorcnt(0);
  }
  __syncthreads();

  const int   nb   = n[row];
  const float p    = param[0];
  const float rho  = tanhf(0.5f * p);           // 2*sigmoid(p) - 1
  const float coef = -rho / (1.0f + rho * rho);

  const int base = tid * ATH_PER_THREAD;
  float d[ATH_PER_THREAD];
  const float4* t4 = reinterpret_cast<const float4*>(&s_yt[base]);
  const float4* p4 = reinterpret_cast<const float4*>(&s_yp[base]);
#pragma unroll
  for (int v = 0; v < ATH_PER_THREAD / 4; ++v) {   // 4x ds_load_b128 per array
    float4 a = t4[v];
    float4 b = p4[v];
    int j = base + 4 * v;
    d[4 * v + 0] = (j + 0 < nb) ? (a.x - b.x) : 0.0f;
    d[4 * v + 1] = (j + 1 < nb) ? (a.y - b.y) : 0.0f;
    d[4 * v + 2] = (j + 2 < nb) ? (a.z - b.z) : 0.0f;
    d[4 * v + 3] = (j + 3 < nb) ? (a.w - b.w) : 0.0f;
  }
  // Neighbor at the chunk edge, straight from LDS (masked; j==D contributes 0).
  float dnext = 0.0f;
  if (base + ATH_PER_THREAD < ATH_D) {
    int j = base + ATH_PER_THREAD;
    float v = s_yt[j] - s_yp[j];
    dnext = (j < nb) ? v : 0.0f;
  }

  // quad contribution: sum d^2 + 2*coef * sum d_j * d_{j+1}
  float s2 = 0.0f, sa = 0.0f;
#pragma unroll
  for (int k = 0; k < ATH_PER_THREAD; ++k) s2 = fmaf(d[k], d[k], s2);
#pragma unroll
  for (int k = 0; k < ATH_PER_THREAD - 1; ++k) sa = fmaf(d[k], d[k + 1], sa);
  sa = fmaf(d[ATH_PER_THREAD - 1], dnext, sa);

  float q = fmaf(2.0f * coef, sa, s2);

  // wave32 tree reduction, then cross-wave via LDS (deterministic order).
#pragma unroll
  for (int off = 16; off > 0; off >>= 1) q += __shfl_xor(q, off, 32);
  if ((tid & 31) == 0) s_red[tid >> 5] = q;
  __syncthreads();
  if (tid == 0) {
    float t = 0.0f;
#pragma unroll
    for (int w = 0; w < ATH_THREADS / 32; ++w) t += s_red[w];
    rowq[row] = t / (float)nb;                 // quad_b / n_b
  }
}

__global__ __launch_bounds__(256) void
ar1_mean_kernel(const float* __restrict__ rowq, float* __restrict__ out, int B) {
  __shared__ float s_red[8];
  float s = 0.0f;
  for (int i = threadIdx.x; i < B; i += 256) s += rowq[i];
#pragma unroll
  for (int off = 16; off > 0; off >>= 1) s += __shfl_xor(s, off, 32);
  if ((threadIdx.x & 31) == 0) s_red[threadIdx.x >> 5] = s;
  __syncthreads();
  if (threadIdx.x == 0) {
    float t = 0.0f;
#pragma unroll
    for (int w = 0; w < 8; ++w) t += s_red[w];
    out[0] = t / (float)B;                     // mean over batch
  }
}

extern "C" void kernel_launch(void* const* d_in, const int* in_sizes, int n_in,
                              void* d_out, int out_size, void* d_ws, size_t ws_size,
                              hipStream_t stream) {
  const float* y_true = (const float*)d_in[0];
  const float* y_pred = (const float*)d_in[1];
  const float* param  = (const float*)d_in[2];
  const int*   n      = (const int*)d_in[3];
  float* out  = (float*)d_out;
  float* rowq = (float*)d_ws;                  // B floats of scratch

  const int B = in_sizes[3];                   // n has one entry per row
  ar1_row_quad_kernel<<<B, ATH_THREADS, 0, stream>>>(y_true, y_pred, param, n, rowq);
  ar1_mean_kernel<<<1, 256, 0, stream>>>(rowq, out, B);
}